// Reduction_Layer_17334488006868
// MI455X (gfx1250) — compile-verified
//
#include <hip/hip_runtime.h>
#include <hip/hip_bf16.h>
#include <stdint.h>

// Problem constants (from the reference)
#define BATCH    64
#define SIZE_IN  4096
#define SIZE_OUT 1024

// Tiling
#define THREADS     256                         // 8 wave32s
#define ROWS_PB     32                          // rows of A per block
#define CHUNK_COLS  128                         // floats per row per chunk
#define NCHUNK      (SIZE_OUT / CHUNK_COLS)     // 8 chunks
#define XFERS_PER_T 4                           // 4 x b128 per thread per chunk

// ---- CDNA5 async global -> LDS copy (16 bytes per lane), ASYNCcnt-tracked ----
// Inline asm form (gfx1250 VGLOBAL async encoding): vdst = LDS byte offset
// VGPR, vaddr = 64-bit global address VGPR pair, no SADDR ("off").
__device__ __forceinline__ void async_b128_to_lds(const void* gptr, void* lptr) {
    // AS3 -> generic lowering is {aperture_hi32, lds_offset32}; the low 32
    // bits are the wave-relative LDS byte address the instruction wants.
    uint32_t loff  = (uint32_t)(uintptr_t)lptr;
    uint64_t gaddr = (uint64_t)(uintptr_t)gptr;
    asm volatile("global_load_async_to_lds_b128 %0, %1, off"
                 :: "v"(loff), "v"(gaddr) : "memory");
}

__device__ __forceinline__ void wait_async_le4() {
#if __has_builtin(__builtin_amdgcn_s_wait_asynccnt)
    __builtin_amdgcn_s_wait_asynccnt(4);
#else
    asm volatile("s_wait_asynccnt 4" ::: "memory");
#endif
}
__device__ __forceinline__ void wait_async_le0() {
#if __has_builtin(__builtin_amdgcn_s_wait_asynccnt)
    __builtin_amdgcn_s_wait_asynccnt(0);
#else
    asm volatile("s_wait_asynccnt 0" ::: "memory");
#endif
}

__device__ __forceinline__ float sigmoidf_fast(float v) {
    return 1.0f / (1.0f + __expf(-v));
}

// out[b,i] = x[b,i] * (x>=0 ? sigmoid(rowmax_k A[i,k]) : sigmoid(rowmin_k A[i,k]))
__global__ __launch_bounds__(THREADS)
void rowminmax_apply_kernel(const float* __restrict__ x,
                            const float* __restrict__ A,
                            float* __restrict__ out) {
    __shared__ __align__(16) float tile[2][ROWS_PB][CHUNK_COLS];  // 2 x 16 KB
    __shared__ float s_wmax[ROWS_PB];
    __shared__ float s_wmin[ROWS_PB];

    const int t    = threadIdx.x;
    const int lane = t & 31;
    const int wave = t >> 5;                  // 0..7
    const int row0 = blockIdx.x * ROWS_PB;    // 128 blocks cover 4096 rows

    // Issue one 16 KB chunk (32 rows x 128 cols) as 1024 async b128 transfers:
    // 4 per thread, 512 B contiguous per wave per instruction.
    auto issue_chunk = [&](int c, int buf) {
        const float* gbase = A + (size_t)row0 * SIZE_OUT + (size_t)c * CHUNK_COLS;
        #pragma unroll
        for (int j = 0; j < XFERS_PER_T; ++j) {
            int e  = j * THREADS + t;         // 0..1023
            int r  = e >> 5;                  // tile row (32 float4 per row)
            int f4 = e & 31;                  // float4 index within row
            async_b128_to_lds(gbase + (size_t)r * SIZE_OUT + f4 * 4,
                              &tile[buf][r][f4 * 4]);
        }
    };

    issue_chunk(0, 0);

    float mn[4], mx[4];
    #pragma unroll
    for (int r = 0; r < 4; ++r) { mn[r] = 3.402823466e38f; mx[r] = -3.402823466e38f; }

    for (int c = 0; c < NCHUNK; ++c) {
        if (c + 1 < NCHUNK) {
            issue_chunk(c + 1, (c + 1) & 1);  // keep next chunk in flight
            wait_async_le4();                 // chunk c complete (async dones in order)
        } else {
            wait_async_le0();
        }
        __syncthreads();                      // chunk c visible to all waves

        // Wave w reduces rows 4w..4w+3; lane reads 16B (banks conflict-free
        // per half-wave: 16 lanes x 16B span all 64 banks exactly once).
        #pragma unroll
        for (int r = 0; r < 4; ++r) {
            const float4 v = *(const float4*)&tile[c & 1][wave * 4 + r][lane * 4];
            mx[r] = fmaxf(mx[r], fmaxf(fmaxf(v.x, v.y), fmaxf(v.z, v.w)));
            mn[r] = fminf(mn[r], fminf(fminf(v.x, v.y), fminf(v.z, v.w)));
        }
        __syncthreads();                      // reads done before buffer reuse
    }

    // Cross-lane butterfly min/max over the wave32.
    #pragma unroll
    for (int off = 16; off >= 1; off >>= 1) {
        #pragma unroll
        for (int r = 0; r < 4; ++r) {
            mx[r] = fmaxf(mx[r], __shfl_xor(mx[r], off, 32));
            mn[r] = fminf(mn[r], __shfl_xor(mn[r], off, 32));
        }
    }
    if (lane == 0) {
        #pragma unroll
        for (int r = 0; r < 4; ++r) {
            // sigmoid is monotone: max_k sigmoid(a) = sigmoid(max_k a)
            s_wmax[wave * 4 + r] = sigmoidf_fast(mx[r]);
            s_wmin[wave * 4 + r] = sigmoidf_fast(mn[r]);
        }
    }
    __syncthreads();

    // Apply: 64 batches x 32 columns = 2048 outputs, 8 per thread, 128 B
    // coalesced segments on both x and out.
    #pragma unroll
    for (int it = 0; it < (BATCH * ROWS_PB) / THREADS; ++it) {
        int e = it * THREADS + t;
        int b = e >> 5;
        int r = e & 31;
        size_t idx = (size_t)b * SIZE_IN + row0 + r;
        float xv = x[idx];
        out[idx] = xv * (xv >= 0.0f ? s_wmax[r] : s_wmin[r]);
    }
}

extern "C" void kernel_launch(void* const* d_in, const int* in_sizes, int n_in,
                              void* d_out, int out_size, void* d_ws, size_t ws_size,
                              hipStream_t stream) {
    const float* x = (const float*)d_in[0];   // (64, 4096) f32
    const float* A = (const float*)d_in[1];   // (4096, 1024) f32
    float* out = (float*)d_out;               // (64, 4096) f32
    (void)in_sizes; (void)n_in; (void)out_size; (void)d_ws; (void)ws_size;

    dim3 grid(SIZE_IN / ROWS_PB);             // 128 blocks
    rowminmax_apply_kernel<<<grid, THREADS, 0, stream>>>(x, A, out);
}